// GCN_27994596836121
// MI455X (gfx1250) — compile-verified
//
#include <hip/hip_runtime.h>
#include <hip/hip_bf16.h>

typedef float v2f __attribute__((ext_vector_type(2)));
typedef float v8f __attribute__((ext_vector_type(8)));

#define F_IN 128
#define HID  128
#define CLS  64

// ---------------------------------------------------------------------------
// Degree accumulation: deg[dst] += 1 for every edge (self-loop handled as +1
// in the rsqrt kernel, since every node gets exactly one self-loop).
// ---------------------------------------------------------------------------
__global__ void gcn_degree_kernel(const int* __restrict__ ei, float* __restrict__ deg, int E) {
    int stride = gridDim.x * blockDim.x;
    for (int e = blockIdx.x * blockDim.x + threadIdx.x; e < E; e += stride) {
        atomicAdd(&deg[ei[E + e]], 1.0f);
    }
}

__global__ void gcn_dinv_kernel(float* __restrict__ deg, int N) {
    int i = blockIdx.x * blockDim.x + threadIdx.x;
    if (i < N) deg[i] = rsqrtf(deg[i] + 1.0f);  // +1 = self loop; always > 0
}

// ---------------------------------------------------------------------------
// FP32 WMMA GEMM: C[M x Nc] = A[M x K] * B[K x Nc]
// One wave per 16x16 output tile, V_WMMA_F32_16X16X4_F32, K stepped by 4.
// M % 16 == 0, Nc % 16 == 0, K % 4 == 0. EXEC is uniform per wave.
// A 16x4 layout: lane = {m = lane&15, half = lane>>4}; K = kbase + 2*half + vgpr
// B 4x16 layout: lane n = lane&15;                     K = kbase + 2*half + vgpr
// C/D: vgpr r, half h -> row r + 8*h, col = lane&15
// ---------------------------------------------------------------------------
__global__ void gcn_wmma_gemm_f32(const float* __restrict__ A,
                                  const float* __restrict__ B,
                                  float* __restrict__ C,
                                  int M, int K, int Nc) {
    int wave = (blockIdx.x * blockDim.x + threadIdx.x) >> 5;
    int lane = threadIdx.x & 31;
    int tilesN = Nc >> 4;
    int tm = wave / tilesN;
    int tn = wave - tm * tilesN;
    if (tm * 16 >= M) return;             // uniform across the wave

    const int m0   = tm << 4;
    const int n0   = tn << 4;
    const int half = lane >> 4;           // 0 or 1
    const int l    = lane & 15;

    v8f acc = {};
    const float* Arow = A + (size_t)(m0 + l) * K;
    for (int k0 = 0; k0 < K; k0 += 4) {
        int ka = k0 + 2 * half;
        v2f a, b;
        a.x = Arow[ka];
        a.y = Arow[ka + 1];
        b.x = B[(size_t)ka       * Nc + n0 + l];
        b.y = B[(size_t)(ka + 1) * Nc + n0 + l];
        acc = __builtin_amdgcn_wmma_f32_16x16x4_f32(
            /*neg_a=*/false, a, /*neg_b=*/false, b,
            /*c_mod=*/(short)0, acc, /*reuse_a=*/false, /*reuse_b=*/false);
    }
#pragma unroll
    for (int r = 0; r < 8; ++r) {
        C[(size_t)(m0 + r + 8 * half) * Nc + n0 + l] = acc[r];
    }
}

// ---------------------------------------------------------------------------
// Layer-1 edge scatter: 32 lanes per edge, float4 (4 features) per lane.
// out[dst] += h1[src] * dinv[src]*dinv[dst]   (accumulated with f32 atomics)
// ---------------------------------------------------------------------------
__global__ void gcn_scatter128_kernel(const int* __restrict__ ei,
                                      const float* __restrict__ dinv,
                                      const float* __restrict__ h,
                                      float* __restrict__ out, int E) {
    int gid  = blockIdx.x * blockDim.x + threadIdx.x;
    int e    = gid >> 5;
    int lane = gid & 31;
    if (e >= E) return;
    int s = ei[e];
    int d = ei[E + e];
    float nrm = dinv[s] * dinv[d];
    const float4* hv = (const float4*)(h + (size_t)s * F_IN);
    float4 v = hv[lane];
    float* o = out + (size_t)d * F_IN + lane * 4;
    atomicAdd(o + 0, v.x * nrm);
    atomicAdd(o + 1, v.y * nrm);
    atomicAdd(o + 2, v.z * nrm);
    atomicAdd(o + 3, v.w * nrm);
}

// Layer-2 edge scatter: 32 lanes per edge, float2 per lane (64 features).
__global__ void gcn_scatter64_kernel(const int* __restrict__ ei,
                                     const float* __restrict__ dinv,
                                     const float* __restrict__ h,
                                     float* __restrict__ out, int E) {
    int gid  = blockIdx.x * blockDim.x + threadIdx.x;
    int e    = gid >> 5;
    int lane = gid & 31;
    if (e >= E) return;
    int s = ei[e];
    int d = ei[E + e];
    float nrm = dinv[s] * dinv[d];
    const float2* hv = (const float2*)(h + (size_t)s * CLS);
    float2 v = hv[lane];
    float* o = out + (size_t)d * CLS + lane * 2;
    atomicAdd(o + 0, v.x * nrm);
    atomicAdd(o + 1, v.y * nrm);
}

// ---------------------------------------------------------------------------
// Layer-1 finalize (in place on emb region of d_out):
//   emb = relu(acc + h1[i]*dinv[i]^2 + b1)   (self-loop folded in here)
// ---------------------------------------------------------------------------
__global__ void gcn_relu_bias_kernel(float* __restrict__ emb,
                                     const float* __restrict__ h1,
                                     const float* __restrict__ dinv,
                                     const float* __restrict__ b1, int N) {
    int idx = blockIdx.x * blockDim.x + threadIdx.x;
    if (idx >= N * F_IN) return;
    int i = idx >> 7;          // / 128
    int c = idx & (F_IN - 1);
    float di = dinv[i];
    float v = emb[idx] + h1[idx] * di * di + b1[c];
    emb[idx] = v > 0.0f ? v : 0.0f;
}

// ---------------------------------------------------------------------------
// Layer-2 finalize + log_softmax, one wave32 per node (2 classes per lane).
// ---------------------------------------------------------------------------
__global__ void gcn_logsoftmax_kernel(float* __restrict__ logits,
                                      const float* __restrict__ h2,
                                      const float* __restrict__ dinv,
                                      const float* __restrict__ b2, int N) {
    int wave = (blockIdx.x * blockDim.x + threadIdx.x) >> 5;
    int lane = threadIdx.x & 31;
    if (wave >= N) return;
    float di   = dinv[wave];
    float self = di * di;
    size_t base = (size_t)wave * CLS;
    float v0 = logits[base + lane]      + h2[base + lane]      * self + b2[lane];
    float v1 = logits[base + 32 + lane] + h2[base + 32 + lane] * self + b2[32 + lane];

    float mx = fmaxf(v0, v1);
#pragma unroll
    for (int off = 16; off > 0; off >>= 1) mx = fmaxf(mx, __shfl_xor(mx, off, 32));
    float s = __expf(v0 - mx) + __expf(v1 - mx);
#pragma unroll
    for (int off = 16; off > 0; off >>= 1) s += __shfl_xor(s, off, 32);
    float lse = mx + __logf(s);

    logits[base + lane]      = v0 - lse;
    logits[base + 32 + lane] = v1 - lse;
}

// ---------------------------------------------------------------------------
// Host-side orchestration (graph-capture safe: only kernels + memsetAsync)
// ---------------------------------------------------------------------------
extern "C" void kernel_launch(void* const* d_in, const int* in_sizes, int n_in,
                              void* d_out, int out_size, void* d_ws, size_t ws_size,
                              hipStream_t stream) {
    const float* x  = (const float*)d_in[0];
    const int*   ei = (const int*)  d_in[1];
    const float* W1 = (const float*)d_in[2];
    const float* b1 = (const float*)d_in[3];
    const float* W2 = (const float*)d_in[4];
    const float* b2 = (const float*)d_in[5];

    const int N = in_sizes[0] / F_IN;   // 100000
    const int E = in_sizes[1] / 2;      // 1600000

    float* logits = (float*)d_out;                       // [N, CLS]
    float* emb    = (float*)d_out + (size_t)N * CLS;     // [N, HID]

    // Workspace layout (256B aligned): dinv | h1 | h2
    char*  wsb  = (char*)d_ws;
    size_t off  = 0;
    float* dinv = (float*)(wsb + off); off += ((size_t)N * 4 + 255) & ~(size_t)255;
    float* h1   = (float*)(wsb + off); off += ((size_t)N * HID * 4 + 255) & ~(size_t)255;
    float* h2   = (float*)(wsb + off);

    // 1. zero accumulators (d_out is poisoned; deg accumulates from 0)
    hipMemsetAsync(d_out, 0, (size_t)out_size * sizeof(float), stream);
    hipMemsetAsync(dinv, 0, (size_t)N * sizeof(float), stream);

    // 2-3. degree + rsqrt normalization
    gcn_degree_kernel<<<2048, 256, 0, stream>>>(ei, dinv, E);
    gcn_dinv_kernel<<<(N + 255) / 256, 256, 0, stream>>>(dinv, N);

    // 4. h1 = x @ W1  (WMMA f32, one wave per 16x16 tile)
    {
        int waves  = (N / 16) * (HID / 16);
        int blocks = (waves + 7) / 8;     // 8 waves / 256-thread block
        gcn_wmma_gemm_f32<<<blocks, 256, 0, stream>>>(x, W1, h1, N, F_IN, HID);
    }

    // 5. scatter layer 1 into emb region
    {
        long long threads = (long long)E * 32;
        int blocks = (int)((threads + 255) / 256);
        gcn_scatter128_kernel<<<blocks, 256, 0, stream>>>(ei, dinv, h1, emb, E);
    }

    // 6. self-loop + bias + relu (in place)
    gcn_relu_bias_kernel<<<(N * F_IN + 255) / 256, 256, 0, stream>>>(emb, h1, dinv, b1, N);

    // 7. h2 = emb @ W2
    {
        int waves  = (N / 16) * (CLS / 16);
        int blocks = (waves + 7) / 8;
        gcn_wmma_gemm_f32<<<blocks, 256, 0, stream>>>(emb, W2, h2, N, HID, CLS);
    }

    // 8. scatter layer 2 into logits region
    {
        long long threads = (long long)E * 32;
        int blocks = (int)((threads + 255) / 256);
        gcn_scatter64_kernel<<<blocks, 256, 0, stream>>>(ei, dinv, h2, logits, E);
    }

    // 9. self-loop + bias + log_softmax (one wave per node)
    {
        long long threads = (long long)N * 32;
        int blocks = (int)((threads + 255) / 256);
        gcn_logsoftmax_kernel<<<blocks, 256, 0, stream>>>(logits, h2, dinv, b2, N);
    }
}